// RuzickaKernel_25718264169371
// MI455X (gfx1250) — compile-verified
//
#include <hip/hip_runtime.h>
#include <cstdint>

// Problem constants (from reference): x:(2048,64), y:(2048,64), out:(2048,1,64)
// out[i,d] = sum_j min(x[i,d],y[j,d]) / (sum_j max(x[i,d],y[j,d]) + eps)
// Identity: sum_j max = M*x + sum_j y_j - sum_j min   (min+max == a+b exactly)
#define RZ_N   2048
#define RZ_M   2048
#define RZ_D   64
#define RZ_EPS 1e-8f

#define TJ      128                 // y rows per LDS tile (32 KB per buffer)
#define NT      (RZ_M / TJ)         // 16 tiles
#define RPT     2                   // x rows per thread
#define ROWS_PB (4 * RPT)           // 8 x rows per block
#define THREADS 256                 // 8 wave32

// Pointee type required by the async-load builtin: 4 x i32 vector.
typedef int v4i __attribute__((vector_size(16)));
typedef __attribute__((address_space(1))) v4i gbl_v4i;   // global (64-bit VA)
typedef __attribute__((address_space(3))) v4i lds_v4i;   // LDS (32-bit offset)

// Issue one 16-byte async global->LDS copy (ASYNCcnt-tracked on CDNA5).
__device__ __forceinline__ void async_copy16(const char* base_uniform,
                                             uint32_t goff, uint32_t lds_byte) {
#if __has_builtin(__builtin_amdgcn_global_load_async_to_lds_b128)
  __builtin_amdgcn_global_load_async_to_lds_b128(
      (gbl_v4i*)(uintptr_t)(base_uniform + goff),   // AS1 source
      (lds_v4i*)lds_byte,                           // AS3 dest (raw byte offset)
      0, 0);                                        // imm offset, cpol
#else
  asm volatile("global_load_async_to_lds_b128 %0, %1, %2 offset:0"
               :
               : "v"(lds_byte), "v"(goff), "s"(base_uniform)
               : "memory");
#endif
}

__device__ __forceinline__ void wait_async0() {
#if __has_builtin(__builtin_amdgcn_s_wait_asynccnt)
  __builtin_amdgcn_s_wait_asynccnt(0);
#else
  asm volatile("s_wait_asynccnt 0" ::: "memory");
#endif
}

__global__ __launch_bounds__(THREADS)
void RuzickaKernel_25718264169371_kernel(const float* __restrict__ x,
                                         const float* __restrict__ y,
                                         float* __restrict__ out) {
  __shared__ float sh[2][TJ * RZ_D];          // double-buffered y tile, 64 KB

  const int tid = threadIdx.x;
  const int d   = tid & (RZ_D - 1);           // 0..63 within a row
  const int il  = tid >> 6;                   // 0..3 local row group
  const int i0  = blockIdx.x * ROWS_PB + il * RPT;   // first x/out row of thread

  // Low 32 bits of a flat pointer to LDS == DS byte offset (ISA flat-aperture rule).
  const uint32_t lds_base = (uint32_t)(uintptr_t)(&sh[0][0]);
  const char*    ybase    = (const char*)y;   // uniform 64-bit base (SGPRs)

  // Stage y tile `t` (TJ rows x 64 floats) into buffer `buf`:
  // 256 threads x 8 rounds x 16 B = 32 KB, fully coalesced.
  auto stage = [&](int buf, int t) {
    const uint32_t tile_bytes = TJ * RZ_D * 4u;                  // 32768
    const uint32_t g0 = (uint32_t)t * tile_bytes;
    const uint32_t l0 = lds_base + (uint32_t)buf * tile_bytes;
#pragma unroll
    for (int r = 0; r < (int)(tile_bytes / (THREADS * 16u)); ++r) {  // 8 rounds
      const uint32_t off = (uint32_t)(r * THREADS + tid) * 16u;
      async_copy16(ybase, g0 + off, l0 + off);
    }
  };

  float xv[RPT], smin[RPT];
#pragma unroll
  for (int r = 0; r < RPT; ++r) {
    xv[r]   = x[(size_t)(i0 + r) * RZ_D + d];
    smin[r] = 0.0f;
  }
  float ssum = 0.0f;                           // sum_j y[j,d], shared by all rows

  stage(0, 0);                                 // prime the pipeline
  for (int t = 0; t < NT; ++t) {
    wait_async0();                             // this wave's async writes landed
    __syncthreads();                           // all waves' portions visible
    if (t + 1 < NT) stage((t + 1) & 1, t + 1); // overlap next tile with compute

    const float* buf = &sh[t & 1][0];
#pragma unroll 8
    for (int j = 0; j < TJ; ++j) {
      const float yv = buf[j * RZ_D + d];      // one ds_load_b32 feeds RPT rows
      ssum += yv;
#pragma unroll
      for (int r = 0; r < RPT; ++r)
        smin[r] += fminf(xv[r], yv);           // v_min_num_f32 + v_add_f32 pairs
    }
    __syncthreads();                           // tile fully consumed before reuse
  }

#pragma unroll
  for (int r = 0; r < RPT; ++r) {
    // sum_j max = M*x + sum_j y - sum_j min  (exact per element; FP reassoc in sums)
    const float den = fmaf((float)RZ_M, xv[r], ssum) - smin[r] + RZ_EPS;
    out[(size_t)(i0 + r) * RZ_D + d] = smin[r] / den;
  }
}

extern "C" void kernel_launch(void* const* d_in, const int* in_sizes, int n_in,
                              void* d_out, int out_size, void* d_ws, size_t ws_size,
                              hipStream_t stream) {
  const float* x = (const float*)d_in[0];
  const float* y = (const float*)d_in[1];
  float* out = (float*)d_out;

  dim3 grid(RZ_N / ROWS_PB);                   // 256 blocks
  dim3 block(THREADS);                         // 8 wave32 each
  RuzickaKernel_25718264169371_kernel<<<grid, block, 0, stream>>>(x, y, out);
}